// Graph_intra_9509057593845
// MI455X (gfx1250) — compile-verified
//
#include <hip/hip_runtime.h>
#include <hip/hip_bf16.h>
#include <climits>

// Problem constants (match reference)
#define N_NODES   50000
#define N_EDGES   800000
#define FIN       128
#define EDGE_DIM  64
#define NHEAD     8
#define CH        32
#define HC        256   // NHEAD * CH
#define NEG_SLOPE 0.2f

typedef __attribute__((ext_vector_type(2))) float v2f;
typedef __attribute__((ext_vector_type(8))) float v8f;

// Monotonic int encoding of IEEE fp32 (finite values): enc(a) < enc(b) <=> a < b
__device__ __forceinline__ int enc_f32(float f) {
    int i = __float_as_int(f);
    return (i >= 0) ? i : (i ^ 0x7fffffff);
}
__device__ __forceinline__ float dec_f32(int i) {
    return __int_as_float((i >= 0) ? i : (i ^ 0x7fffffff));
}

// out[n, j] = bias[j]  (scatter-add target; bias folded in up front)
__global__ void k_init_out(const float* __restrict__ bias, float* __restrict__ out) {
    int t = blockIdx.x * blockDim.x + threadIdx.x;
    if (t >= N_NODES * HC) return;
    out[t] = bias[t & (HC - 1)];
}

// seg_max encoded sentinel + seg_sum zero
__global__ void k_init_seg(int* __restrict__ seg_max_enc, float* __restrict__ seg_sum) {
    int t = blockIdx.x * blockDim.x + threadIdx.x;
    if (t >= N_NODES * NHEAD) return;
    seg_max_enc[t] = INT_MIN;
    seg_sum[t] = 0.0f;
}

// w_e[k, h] = sum_c W_edge[k, h*32 + c] * att_edge[h, c]
// Folds the E x 64 x 256 GEMM + einsum into an E x 64 x 8 GEMV.
__global__ void k_prep_we(const float* __restrict__ W_edge,
                          const float* __restrict__ att_edge,
                          float* __restrict__ w_e) {
    int t = blockIdx.x * blockDim.x + threadIdx.x;
    if (t >= EDGE_DIM * NHEAD) return;
    int k = t >> 3, h = t & 7;
    const float* wr = W_edge + k * HC + h * CH;
    const float* ar = att_edge + h * CH;
    float acc = 0.0f;
    #pragma unroll
    for (int c = 0; c < CH; ++c) acc += wr[c] * ar[c];
    w_e[k * NHEAD + h] = acc;
}

// xs = relu(x) @ W via V_WMMA_F32_16X16X4_F32 (exact fp32).
// One wave (32 lanes) computes one 16x16 output tile, K-loop = 128/4.
// A 16x4 f32 layout (ISA 7.12.2): lanes 0-15 -> M, VGPR0/1 = K0/K1; lanes 16-31 = K2/K3.
// B 4x16 f32 layout (symmetric):  lanes 0-15 -> N, VGPR0/1 = K0/K1; lanes 16-31 = K2/K3.
// C/D 16x16 f32: VGPR r holds row m0+r (lanes 0-15) and m0+r+8 (lanes 16-31).
__global__ void k_gemm_wmma(const float* __restrict__ x,
                            const float* __restrict__ W,
                            float* __restrict__ xs) {
    const int lane = threadIdx.x & 31;
    const int half = lane >> 4;     // 0: K+{0,1}, 1: K+{2,3}
    const int l    = lane & 15;
    const int m0   = blockIdx.y << 4;
    const int n0   = blockIdx.x << 4;

    const float* arow = x + (size_t)(m0 + l) * FIN;   // A row for this lane
    const float* bcol = W + n0 + l;                   // B column for this lane

    v8f acc = {0.f, 0.f, 0.f, 0.f, 0.f, 0.f, 0.f, 0.f};
    for (int k = 0; k < FIN; k += 4) {
        const int ka = k + half * 2;
        v2f a, b;
        a[0] = fmaxf(arow[ka],     0.0f);             // fused ReLU on A
        a[1] = fmaxf(arow[ka + 1], 0.0f);
        b[0] = bcol[(size_t)ka * HC];
        b[1] = bcol[(size_t)(ka + 1) * HC];
        acc = __builtin_amdgcn_wmma_f32_16x16x4_f32(
            /*neg_a=*/false, a, /*neg_b=*/false, b,
            /*c_mod=*/(short)0, acc, /*reuse_a=*/false, /*reuse_b=*/false);
    }
    #pragma unroll
    for (int r = 0; r < 8; ++r) {
        const int row = m0 + r + half * 8;
        xs[(size_t)row * HC + n0 + l] = acc[r];
    }
}

// a_src[n,h], a_dst[n,h] = <xs[n,h,:], att_src/att_dst[h,:]>
__global__ void k_node_logits(const float* __restrict__ xs,
                              const float* __restrict__ att_src,
                              const float* __restrict__ att_dst,
                              float* __restrict__ a_src,
                              float* __restrict__ a_dst) {
    int t = blockIdx.x * blockDim.x + threadIdx.x;
    if (t >= N_NODES * NHEAD) return;
    int n = t >> 3, h = t & 7;
    const float* xp = xs + (size_t)n * HC + h * CH;
    const float* as = att_src + h * CH;
    const float* ad = att_dst + h * CH;
    float s = 0.f, d = 0.f;
    #pragma unroll
    for (int c = 0; c < CH; ++c) { float v = xp[c]; s += v * as[c]; d += v * ad[c]; }
    a_src[t] = s;
    a_dst[t] = d;
}

// Per-edge logits: a_edge via w_e, add gathered node logits, LeakyReLU,
// and atomic segment-max (int-encoded fp32 -> GLOBAL_ATOMIC_MAX_I32).
__global__ void k_edge_logits(const float* __restrict__ edge_attr,
                              const int* __restrict__ srcI,
                              const int* __restrict__ dstI,
                              const float* __restrict__ w_e,
                              const float* __restrict__ a_src,
                              const float* __restrict__ a_dst,
                              float* __restrict__ alpha,
                              int* __restrict__ seg_max_enc) {
    int t = blockIdx.x * blockDim.x + threadIdx.x;
    if (t >= N_EDGES * NHEAD) return;
    int e = t >> 3, h = t & 7;
    const float* ea = edge_attr + (size_t)e * EDGE_DIM;
    float acc = 0.f;
    #pragma unroll 8
    for (int k = 0; k < EDGE_DIM; ++k) acc += ea[k] * w_e[k * NHEAD + h];
    int s = srcI[e], d = dstI[e];
    float v = a_src[s * NHEAD + h] + a_dst[d * NHEAD + h] + acc;
    v = (v > 0.f) ? v : NEG_SLOPE * v;   // LeakyReLU
    alpha[t] = v;
    atomicMax(&seg_max_enc[d * NHEAD + h], enc_f32(v));
}

// alpha <- exp(alpha - seg_max[dst]); seg_sum[dst] += alpha
__global__ void k_edge_exp(const int* __restrict__ dstI,
                           const int* __restrict__ seg_max_enc,
                           float* __restrict__ alpha,
                           float* __restrict__ seg_sum) {
    int t = blockIdx.x * blockDim.x + threadIdx.x;
    if (t >= N_EDGES * NHEAD) return;
    int e = t >> 3, h = t & 7;
    int d = dstI[e];
    float m = dec_f32(seg_max_enc[d * NHEAD + h]);
    float a = expf(alpha[t] - m);
    alpha[t] = a;
    atomicAdd(&seg_sum[d * NHEAD + h], a);
}

// One 256-thread block per edge: wave w handles head h=w (alpha/seg_sum become
// wave-uniform scalars), coalesced 128B gather of xs[src] and coalesced
// global_atomic_add_f32 into out[dst] (resolves in L2: both arrays < 192MB L2).
__global__ void k_aggregate(const int* __restrict__ srcI,
                            const int* __restrict__ dstI,
                            const float* __restrict__ xs,
                            const float* __restrict__ alpha,
                            const float* __restrict__ seg_sum,
                            float* __restrict__ out) {
    int e = blockIdx.x;
    int j = threadIdx.x;          // 0..255
    int h = j >> 5;               // wave-uniform
    int s = srcI[e], d = dstI[e];
    float w = alpha[e * NHEAD + h] / (seg_sum[d * NHEAD + h] + 1e-16f);
    atomicAdd(&out[(size_t)d * HC + j], w * xs[(size_t)s * HC + j]);
}

extern "C" void kernel_launch(void* const* d_in, const int* in_sizes, int n_in,
                              void* d_out, int out_size, void* d_ws, size_t ws_size,
                              hipStream_t stream) {
    const float* x         = (const float*)d_in[0];
    const int*   edge_idx  = (const int*)  d_in[1];   // [2, E]
    const float* edge_attr = (const float*)d_in[2];
    const float* W         = (const float*)d_in[3];
    const float* att_src   = (const float*)d_in[4];
    const float* att_dst   = (const float*)d_in[5];
    const float* W_edge    = (const float*)d_in[6];
    const float* att_edge  = (const float*)d_in[7];
    const float* bias      = (const float*)d_in[8];
    float* out = (float*)d_out;

    const int* srcI = edge_idx;
    const int* dstI = edge_idx + N_EDGES;

    // Workspace carve-out (~83 MB, all fp32/int32, 4B aligned)
    float* ws          = (float*)d_ws;
    float* xs          = ws;                                        // N*HC
    float* a_src       = xs + (size_t)N_NODES * HC;                 // N*H
    float* a_dst       = a_src + (size_t)N_NODES * NHEAD;           // N*H
    float* alpha       = a_dst + (size_t)N_NODES * NHEAD;           // E*H
    float* seg_sum     = alpha + (size_t)N_EDGES * NHEAD;           // N*H
    int*   seg_max_enc = (int*)(seg_sum + (size_t)N_NODES * NHEAD); // N*H
    float* w_e         = (float*)(seg_max_enc + (size_t)N_NODES * NHEAD); // 64*8

    k_init_out<<<(N_NODES * HC + 255) / 256, 256, 0, stream>>>(bias, out);
    k_init_seg<<<(N_NODES * NHEAD + 255) / 256, 256, 0, stream>>>(seg_max_enc, seg_sum);
    k_prep_we<<<(EDGE_DIM * NHEAD + 63) / 64, 64, 0, stream>>>(W_edge, att_edge, w_e);

    dim3 ggemm(HC / 16, N_NODES / 16);   // 16 x 3125 tiles, 1 wave per tile
    k_gemm_wmma<<<ggemm, 32, 0, stream>>>(x, W, xs);

    k_node_logits<<<(N_NODES * NHEAD + 255) / 256, 256, 0, stream>>>(
        xs, att_src, att_dst, a_src, a_dst);
    k_edge_logits<<<(N_EDGES * NHEAD + 255) / 256, 256, 0, stream>>>(
        edge_attr, srcI, dstI, w_e, a_src, a_dst, alpha, seg_max_enc);
    k_edge_exp<<<(N_EDGES * NHEAD + 255) / 256, 256, 0, stream>>>(
        dstI, seg_max_enc, alpha, seg_sum);
    k_aggregate<<<N_EDGES, HC, 0, stream>>>(srcI, dstI, xs, alpha, seg_sum, out);
}